// SEGNNMessage_30915174596961
// MI455X (gfx1250) — compile-verified
//
#include <hip/hip_runtime.h>
#include <hip/hip_bf16.h>

typedef __attribute__((ext_vector_type(2))) float v2f;
typedef __attribute__((ext_vector_type(8))) float v8f;

#define LDS_STRIDE 132   // 132 mod 64 = 4 -> conflict-free fragment reads

__device__ __forceinline__ v8f wmma_f32(v2f a, v2f b, v8f c) {
    // V_WMMA_F32_16X16X4_F32 : D(16x16,f32) = A(16x4,f32) x B(4x16,f32) + C
    return __builtin_amdgcn_wmma_f32_16x16x4_f32(
        false, a, false, b, (short)0, c, false, false);
}

__device__ __forceinline__ float fsigmoid(float x) {
    return 1.0f / (1.0f + __expf(-x));
}

// Repack row-major W[K][Ncols] into B-fragment order, folding in `scale`.
// Fragment (kk, nt), lane L, pair p: W[4*kk + 2*(L>>4) + p][16*nt + (L&15)]
__global__ void pack_b_kernel(const float* __restrict__ W, float* __restrict__ out,
                              int K, int Ncols, float scale) {
    int tid = blockIdx.x * blockDim.x + threadIdx.x;
    int total = K * Ncols;
    if (tid >= total) return;
    int p    = tid & 1;
    int L    = (tid >> 1) & 31;
    int frag = tid >> 6;
    int NT   = Ncols >> 4;
    int nt   = frag % NT;
    int kk   = frag / NT;
    int row  = 4 * kk + 2 * (L >> 4) + p;
    int col  = 16 * nt + (L & 15);
    out[tid] = W[row * Ncols + col] * scale;
}

// Per-node transform: xt[n][0..31] = (s_in @ W1_s)*inv_sqrt_mul (folded in pack)
//                     xt[n][32+32*i+w] = sum_u v_in[n,u,i]*W1_v[u,w]  (planar by i)
__global__ void node_transform_kernel(const float* __restrict__ nf,
                                      const float* __restrict__ pW1s,
                                      const float* __restrict__ pW1v,
                                      float* __restrict__ xt, int N) {
    int wave = (blockIdx.x * blockDim.x + threadIdx.x) >> 5;
    int lane = threadIdx.x & 31;
    int tile = wave;
    if (tile * 16 >= N) return;
    int j = lane & 15, h = lane >> 4;
    bool full = (tile * 16 + 16 <= N);            // uniform across wave
    int na = min(tile * 16 + j, N - 1);
    const float* arow = nf + (size_t)na * 128;

    v8f acc[8];   // [channel tile]: 0..1 = s cols 0..31; 2..7 = v planar i, cols per nt
    #pragma unroll
    for (int t = 0; t < 8; ++t) acc[t] = v8f{};

    // scalar channel: (16x32) @ (32x32)
    for (int kk = 0; kk < 8; ++kk) {
        int k0 = 4 * kk + 2 * h;
        v2f a; a.x = arow[k0]; a.y = arow[k0 + 1];
        v2f b0 = *(const v2f*)(pW1s + ((size_t)(kk * 2 + 0) * 32 + lane) * 2);
        v2f b1 = *(const v2f*)(pW1s + ((size_t)(kk * 2 + 1) * 32 + lane) * 2);
        acc[0] = wmma_f32(a, b0, acc[0]);
        acc[1] = wmma_f32(a, b1, acc[1]);
    }
    // vector channels: per component i, (16x32) @ (32x32); input col = 32 + 3*u + i
    for (int i = 0; i < 3; ++i) {
        for (int kk = 0; kk < 8; ++kk) {
            int k0 = 4 * kk + 2 * h;
            v2f a; a.x = arow[32 + 3 * k0 + i]; a.y = arow[32 + 3 * (k0 + 1) + i];
            v2f b0 = *(const v2f*)(pW1v + ((size_t)(kk * 2 + 0) * 32 + lane) * 2);
            v2f b1 = *(const v2f*)(pW1v + ((size_t)(kk * 2 + 1) * 32 + lane) * 2);
            acc[2 + 2 * i]     = wmma_f32(a, b0, acc[2 + 2 * i]);
            acc[2 + 2 * i + 1] = wmma_f32(a, b1, acc[2 + 2 * i + 1]);
        }
    }

    if (full) {
        #pragma unroll
        for (int r = 0; r < 8; ++r) {
            float* xrow = xt + (size_t)(tile * 16 + r + 8 * h) * 128;
            xrow[j]      = acc[0][r];
            xrow[16 + j] = acc[1][r];
            #pragma unroll
            for (int i = 0; i < 3; ++i) {
                xrow[32 + 32 * i + j]      = acc[2 + 2 * i][r];
                xrow[32 + 32 * i + 16 + j] = acc[2 + 2 * i + 1][r];
            }
        }
    } else {
        for (int r = 0; r < 8; ++r) {
            int nd = tile * 16 + r + 8 * h;
            if (nd >= N) continue;
            float* xrow = xt + (size_t)nd * 128;
            xrow[j]      = acc[0][r];
            xrow[16 + j] = acc[1][r];
            for (int i = 0; i < 3; ++i) {
                xrow[32 + 32 * i + j]      = acc[2 + 2 * i][r];
                xrow[32 + 32 * i + 16 + j] = acc[2 + 2 * i + 1][r];
            }
        }
    }
}

__global__ void __launch_bounds__(128)
edge_kernel(const float* __restrict__ xt,
            const int*   __restrict__ edge_index,
            const float* __restrict__ edge_attrs,
            const float* __restrict__ edge_emb,
            const float* __restrict__ pW0,
            const float* __restrict__ pW1,
            const float* __restrict__ pW2s,
            const float* __restrict__ pW2v,
            float* __restrict__ out, int E) {
    __shared__ float lds[4][16 * LDS_STRIDE];   // per-wave 16x(<=128) staging, padded
    const float INV_SQRT3 = 0.57735026918962576f;

    int wid  = threadIdx.x >> 5;
    int lane = threadIdx.x & 31;
    int tile = blockIdx.x * 4 + wid;
    if (tile * 16 >= E) return;
    bool full = (tile * 16 + 16 <= E);          // uniform across wave
    float* wl = lds[wid];
    int j = lane & 15, h = lane >> 4;

    int ej  = min(tile * 16 + j, E - 1);
    int src = edge_index[ej];                       // row 0 of (2,E)
    const float* base = xt + (size_t)src * 128;     // transformed node row (L2-resident)
    const float4 ea = *(const float4*)(edge_attrs + (size_t)ej * 4);
    float y0 = ea.x, y1a = ea.y, y1b = ea.z, y1c = ea.w;
    const float* emb = edge_emb + (size_t)ej * 8;

    // ---- MLP stage 1: hmid = silu(emb(16x8) @ mlp_w0(8x64)) ----
    v8f hacc[4] = {};
    for (int kk = 0; kk < 2; ++kk) {
        int k0 = 4 * kk + 2 * h;
        v2f a = *(const v2f*)(emb + k0);
        for (int nt = 0; nt < 4; ++nt) {
            v2f b = *(const v2f*)(pW0 + ((size_t)(kk * 4 + nt) * 32 + lane) * 2);
            hacc[nt] = wmma_f32(a, b, hacc[nt]);
        }
    }
    for (int nt = 0; nt < 4; ++nt)
        for (int r = 0; r < 8; ++r) {
            float x = hacc[nt][r];
            wl[(r + 8 * h) * LDS_STRIDE + 16 * nt + j] = x * fsigmoid(x);
        }

    // ---- MLP stage 2: w = hmid(16x64) @ mlp_w1(64x128) ----
    const float* myrow = wl + j * LDS_STRIDE;
    v8f wacc[8] = {};
    for (int kk = 0; kk < 16; ++kk) {
        int k0 = 4 * kk + 2 * h;
        v2f a = *(const v2f*)(myrow + k0);
        for (int nt = 0; nt < 8; ++nt) {
            v2f b = *(const v2f*)(pW1 + ((size_t)(kk * 8 + nt) * 32 + lane) * 2);
            wacc[nt] = wmma_f32(a, b, wacc[nt]);
        }
    }
    // stage w (16x128) to LDS (same slice; per-wave DS ops are in-order)
    for (int nt = 0; nt < 8; ++nt)
        for (int r = 0; r < 8; ++r)
            wl[(r + 8 * h) * LDS_STRIDE + 16 * nt + j] = wacc[nt][r];

    // ---- tensor product + output GEMMs, A-fragments built on the fly ----
    v8f msacc[4] = {};
    v8f mvacc[3][2] = {};
    for (int kk = 0; kk < 16; ++kk) {
        int k0 = 4 * kk + 2 * h;
        v2f tsA, tvA0, tvA1, tvA2;
        if (kk < 8) {   // tp cols [0,32): p1 = w1*se*y0 ; p2[i] = w2*se*y1[i]
            int u0 = k0;
            v2f w1p = *(const v2f*)(myrow + u0);
            v2f w2p = *(const v2f*)(myrow + 64 + u0);
            v2f sep; sep.x = base[u0]; sep.y = base[u0 + 1];
            float a0 = w2p.x * sep.x, a1 = w2p.y * sep.y;
            tsA.x  = w1p.x * sep.x * y0;  tsA.y  = w1p.y * sep.y * y0;
            tvA0.x = a0 * y1a;            tvA0.y = a1 * y1a;
            tvA1.x = a0 * y1b;            tvA1.y = a1 * y1b;
            tvA2.x = a0 * y1c;            tvA2.y = a1 * y1c;
        } else {        // tp cols [32,64): p4 = w4*(ve.y1)/sqrt3 ; p3[i] = w3*ve[i]*y0
            int u0 = k0 - 32;
            v2f w4p = *(const v2f*)(myrow + 32 + u0);
            v2f w3p = *(const v2f*)(myrow + 96 + u0);
            v2f v0p = *(const v2f*)(base + 32 + u0);
            v2f v1p = *(const v2f*)(base + 64 + u0);
            v2f v2p = *(const v2f*)(base + 96 + u0);
            float d0 = v0p.x * y1a + v1p.x * y1b + v2p.x * y1c;
            float d1 = v0p.y * y1a + v1p.y * y1b + v2p.y * y1c;
            tsA.x  = w4p.x * d0 * INV_SQRT3;  tsA.y  = w4p.y * d1 * INV_SQRT3;
            tvA0.x = w3p.x * v0p.x * y0;      tvA0.y = w3p.y * v0p.y * y0;
            tvA1.x = w3p.x * v1p.x * y0;      tvA1.y = w3p.y * v1p.y * y0;
            tvA2.x = w3p.x * v2p.x * y0;      tvA2.y = w3p.y * v2p.y * y0;
        }
        for (int nt = 0; nt < 4; ++nt) {
            v2f b = *(const v2f*)(pW2s + ((size_t)(kk * 4 + nt) * 32 + lane) * 2);
            msacc[nt] = wmma_f32(tsA, b, msacc[nt]);
        }
        for (int nt = 0; nt < 2; ++nt) {
            v2f b = *(const v2f*)(pW2v + ((size_t)(kk * 2 + nt) * 32 + lane) * 2);
            mvacc[0][nt] = wmma_f32(tvA0, b, mvacc[0][nt]);
            mvacc[1][nt] = wmma_f32(tvA1, b, mvacc[1][nt]);
            mvacc[2][nt] = wmma_f32(tvA2, b, mvacc[2][nt]);
        }
    }

    // ---- gating + store ----
    // out[e][0..31] = silu(ms[:32]); out[e][32+3u+i] = sigmoid(ms[32+u]) * mv[u,i]
    if (full) {
        #pragma unroll
        for (int r = 0; r < 8; ++r) {
            float* orow = out + (size_t)(tile * 16 + r + 8 * h) * 128;
            #pragma unroll
            for (int nt = 0; nt < 2; ++nt) {
                float x = msacc[nt][r];
                orow[16 * nt + j] = x * fsigmoid(x);
            }
            #pragma unroll
            for (int nt = 0; nt < 2; ++nt) {
                float g = fsigmoid(msacc[2 + nt][r]);
                int u = 16 * nt + j;
                orow[32 + 3 * u + 0] = g * mvacc[0][nt][r];   // contiguous triple ->
                orow[32 + 3 * u + 1] = g * mvacc[1][nt][r];   //   mergeable store
                orow[32 + 3 * u + 2] = g * mvacc[2][nt][r];
            }
        }
    } else {
        for (int r = 0; r < 8; ++r) {
            int ed = tile * 16 + r + 8 * h;
            if (ed >= E) continue;
            float* orow = out + (size_t)ed * 128;
            for (int nt = 0; nt < 2; ++nt) {
                float x = msacc[nt][r];
                orow[16 * nt + j] = x * fsigmoid(x);
            }
            for (int nt = 0; nt < 2; ++nt) {
                float g = fsigmoid(msacc[2 + nt][r]);
                int u = 16 * nt + j;
                orow[32 + 3 * u + 0] = g * mvacc[0][nt][r];
                orow[32 + 3 * u + 1] = g * mvacc[1][nt][r];
                orow[32 + 3 * u + 2] = g * mvacc[2][nt][r];
            }
        }
    }
}

extern "C" void kernel_launch(void* const* d_in, const int* in_sizes, int n_in,
                              void* d_out, int out_size, void* d_ws, size_t ws_size,
                              hipStream_t stream) {
    const float* node_feats = (const float*)d_in[0];
    const float* edge_attrs = (const float*)d_in[1];
    const float* edge_emb   = (const float*)d_in[2];
    const int*   edge_index = (const int*)  d_in[3];
    const float* W1_s   = (const float*)d_in[4];
    const float* W1_v   = (const float*)d_in[5];
    const float* mlp_w0 = (const float*)d_in[6];
    const float* mlp_w1 = (const float*)d_in[7];
    const float* W2_s   = (const float*)d_in[8];
    const float* W2_v   = (const float*)d_in[9];

    int N = in_sizes[0] / 128;   // 4*MUL
    int E = in_sizes[1] / 4;

    float* ws = (float*)d_ws;
    float* xt = ws;                      // transformed nodes: N x 128
    size_t off = (size_t)N * 128;
    float* pW1s = ws + off; off += 32 * 32;
    float* pW1v = ws + off; off += 32 * 32;
    float* pW0  = ws + off; off += 8 * 64;
    float* pW1  = ws + off; off += 64 * 128;
    float* pW2s = ws + off; off += 64 * 64;
    float* pW2v = ws + off; off += 64 * 32;

    const float inv_sqrt_mul  = 0.17677669529663687f;  // 1/sqrt(32)
    const float inv_sqrt_2mul = 0.125f;                // 1/sqrt(64)

    pack_b_kernel<<<(32 * 32 + 255) / 256, 256, 0, stream>>>(W1_s,   pW1s, 32,  32, inv_sqrt_mul);
    pack_b_kernel<<<(32 * 32 + 255) / 256, 256, 0, stream>>>(W1_v,   pW1v, 32,  32, inv_sqrt_mul);
    pack_b_kernel<<<(8 * 64 + 255) / 256, 256, 0, stream>>>(mlp_w0, pW0,   8,  64, 1.0f);
    pack_b_kernel<<<(64 * 128 + 255) / 256, 256, 0, stream>>>(mlp_w1, pW1,  64, 128, 1.0f);
    pack_b_kernel<<<(64 * 64 + 255) / 256, 256, 0, stream>>>(W2_s,   pW2s, 64,  64, inv_sqrt_2mul);
    pack_b_kernel<<<(64 * 32 + 255) / 256, 256, 0, stream>>>(W2_v,   pW2v, 64,  32, inv_sqrt_2mul);

    int ntiles = (N + 15) / 16;
    node_transform_kernel<<<(ntiles + 3) / 4, 128, 0, stream>>>(node_feats, pW1s, pW1v, xt, N);

    int etiles = (E + 15) / 16;
    edge_kernel<<<(etiles + 3) / 4, 128, 0, stream>>>(xt, edge_index, edge_attrs, edge_emb,
                                                      pW0, pW1, pW2s, pW2v,
                                                      (float*)d_out, E);
}